// Conditioning_45595372815055
// MI455X (gfx1250) — compile-verified
//
#include <hip/hip_runtime.h>
#include <hip/hip_bf16.h>
#include <stdint.h>

// Conditioning: out = concat(lc, gather(W^T, ids) + b) along channel dim.
// Pure streaming kernel (~405 MB of traffic, ~17us at 23.3 TB/s). Zero useful
// FLOPs -> WMMA is the wrong tool (one-hot matmul is multiply-by-zero waste).
// CDNA5 path used: per-lane async global->LDS DMA (free transpose of W into
// the gather table) + s_wait_asynccnt, then 128-bit nontemporal streams.

typedef float v4f __attribute__((ext_vector_type(4)));

#define NSPK 128
#define NEMB 64
#define TBL_ELEMS (NSPK * NEMB)   // 8192 floats = 32 KB LDS
#define BLOCK 256
#define GRID 2048

__global__ __launch_bounds__(BLOCK)
void Conditioning_45595372815055_kernel(const float* __restrict__ lc,
                                        const int*   __restrict__ ids,
                                        const float* __restrict__ W,   // (NEMB, NSPK) row-major
                                        const float* __restrict__ b,   // (NEMB)
                                        float* __restrict__ out,       // (n_pos, NEMB + NEMB... actually I+NEMB=128)
                                        int n_pos)                     // B*T
{
    // table[s*NEMB + e] = W[e*NSPK + s]  (i.e. W^T, rows contiguous for the gather)
    __shared__ float table[TBL_ELEMS];

    const int tid = threadIdx.x;

    // ---- Stage W^T into LDS via gfx1250 async global->LDS DMA.
    // Each lane supplies its own global byte offset and its own LDS byte
    // address, so the transpose happens inside the DMA for free.
    {
        uint32_t lds_base = (uint32_t)(uintptr_t)(&table[0]);  // flat shared addr: low 32 bits = LDS offset
        const int e = tid & (NEMB - 1);                        // constant per thread (stride 256 is mult. of 64)
        #pragma unroll
        for (int k = 0; k < TBL_ELEMS / BLOCK; ++k) {
            int i = k * BLOCK + tid;                 // table element index; s = i>>6, e = i&63
            int s = i >> 6;
            uint32_t voff  = (uint32_t)((e * NSPK + s) * 4);   // byte offset into W
            uint32_t laddr = lds_base + (uint32_t)(i * 4);     // LDS byte address
            asm volatile("global_load_async_to_lds_b32 %0, %1, %2"
                         :: "v"(laddr), "v"(voff), "s"(W)
                         : "memory");
        }
        asm volatile("s_wait_asynccnt 0" ::: "memory");
    }
    __syncthreads();

    // Per-thread loop-invariant channel quad and its bias (j strides are
    // multiples of 16, so c4 == tid & 15 for every iteration).
    const int c4 = tid & 15;
    const v4f biasv = *(const v4f*)(b + c4 * 4);

    const v4f* __restrict__ lc4  = (const v4f*)lc;
    const v4f* __restrict__ tbl4 = (const v4f*)table;
    v4f* __restrict__       out4 = (v4f*)out;

    const long long n_j    = (long long)n_pos * 16;                   // one float4-pair per j
    const long long stride = (long long)gridDim.x * (long long)blockDim.x;
    for (long long j = (long long)blockIdx.x * BLOCK + tid; j < n_j; j += stride) {
        long long pos = j >> 4;
        int id = ids[pos];                                            // broadcast across 16 lanes
        v4f lcv = __builtin_nontemporal_load(lc4 + pos * 16 + c4);    // streaming read
        v4f gcv = tbl4[id * 16 + c4] + biasv;                         // conflict-free ds_load_b128
        __builtin_nontemporal_store(lcv, out4 + pos * 32 + c4);       // out[pos][0:64]   = lc
        __builtin_nontemporal_store(gcv, out4 + pos * 32 + 16 + c4);  // out[pos][64:128] = gc
    }
}

extern "C" void kernel_launch(void* const* d_in, const int* in_sizes, int n_in,
                              void* d_out, int out_size, void* d_ws, size_t ws_size,
                              hipStream_t stream) {
    const float* lc  = (const float*)d_in[0];
    const int*   ids = (const int*)d_in[1];
    const float* W   = (const float*)d_in[2];
    const float* b   = (const float*)d_in[3];
    float*       out = (float*)d_out;
    const int n_pos  = in_sizes[1];   // B*T = element count of ids

    Conditioning_45595372815055_kernel<<<GRID, BLOCK, 0, stream>>>(lc, ids, W, b, out, n_pos);
}